// MultiHeadAttentionPooling_45973329936790
// MI455X (gfx1250) — compile-verified
//
#include <hip/hip_runtime.h>

// ---------------------------------------------------------------------------
// MultiHeadAttentionPooling, restructured for MI455X (gfx1250, wave32, WMMA).
//
// N=262144 atoms, D=256 in, H=8 heads, Dh=32, B=2048 graphs.
//   scores = x @ wq + cq            (wq = (Wk·q)/sqrt(Dh),  [N,256]@[256,8])
//   w      = segment_softmax(scores)
//   xw[b,h,:] = sum_n w[n,h] x[n,:] (weighted segment sum, sum(w)=1 per seg)
//   out    = xw_flat @ Wcomb + bout (Wcomb = blockdiag(Wv)^T Wo, bout=bo+bv@Wo)
// Memory-bound: 2 streams of x (2 x 268 MB).  GEMMs use v_wmma_f32_16x16x4_f32.
// ---------------------------------------------------------------------------

typedef float v2f __attribute__((ext_vector_type(2)));
typedef float v8f __attribute__((ext_vector_type(8)));

#define NATOMS   262144
#define IND      256
#define NHEADS   8
#define HEADD    32
#define NGRAPHS  2048
#define KCOMB    2048            // H * IND
#define INV_SCALE 0.17677669529663687f   // 1/sqrt(32)

// -------------------------- prep: wq, cq, bout ------------------------------
__global__ void mhap_prep(const float* __restrict__ q,   // [8][32]
                          const float* __restrict__ Wk,  // [256][256]
                          const float* __restrict__ bk,  // [256]
                          const float* __restrict__ bv,  // [256]
                          const float* __restrict__ Wo,  // [256][256]
                          const float* __restrict__ bo,  // [256]
                          float* __restrict__ wq,        // [256][16] (cols 8..15 = 0)
                          float* __restrict__ cq,        // [16]
                          float* __restrict__ bout) {    // [256]
    int i = threadIdx.x;  // 0..255
    // wq[i][h] = (sum_d Wk[i][h*32+d] * q[h][d]) / SCALE
    for (int hh = 0; hh < 16; ++hh) {
        float s = 0.0f;
        if (hh < 8) {
            for (int d = 0; d < HEADD; ++d)
                s = fmaf(Wk[i * IND + hh * HEADD + d], q[hh * HEADD + d], s);
            s *= INV_SCALE;
        }
        wq[i * 16 + hh] = s;
    }
    if (i < 16) {
        float s = 0.0f;
        if (i < 8) {
            for (int d = 0; d < HEADD; ++d)
                s = fmaf(bk[i * HEADD + d], q[i * HEADD + d], s);
            s *= INV_SCALE;
        }
        cq[i] = s;
    }
    // bout[j] = bo[j] + sum_k bv[k] * Wo[k][j]
    float s = bo[i];
    for (int k = 0; k < IND; ++k) s = fmaf(bv[k], Wo[k * IND + i], s);
    bout[i] = s;
}

// ---------------- Wcomb[h*256+i][j] = sum_d Wv[i][h*32+d] * Wo[h*32+d][j] ---
__global__ void mhap_wcomb(const float* __restrict__ Wv,
                           const float* __restrict__ Wo,
                           float* __restrict__ wcomb) {  // [2048][256]
    int k = blockIdx.x;       // 0..2047
    int j = threadIdx.x;      // 0..255
    int h = k >> 8, i = k & 255;
    float s = 0.0f;
#pragma unroll
    for (int d = 0; d < HEADD; ++d)
        s = fmaf(Wv[i * IND + h * HEADD + d], Wo[(h * HEADD + d) * IND + j], s);
    wcomb[k * IND + j] = s;
}

// ------------------- scores = x @ wq + cq  (WMMA f32 16x16x4) ---------------
// Block = 256 threads = 8 waves, each wave owns a 16-row tile. 2048 blocks.
__global__ void mhap_scores(const float* __restrict__ x,     // [N][256]
                            const float* __restrict__ wq,    // [256][16]
                            const float* __restrict__ cq,    // [16]
                            float* __restrict__ sw) {        // [N][8]
    __shared__ float lwq[IND * 16];
    for (int t = threadIdx.x; t < IND * 16; t += 256) lwq[t] = wq[t];
    __syncthreads();

    const int wave = threadIdx.x >> 5;
    const int lane = threadIdx.x & 31;
    const int m    = lane & 15;    // A row within tile; also B/C column
    const int kg   = lane >> 4;    // K-group (0/1)

    const long row0 = ((long)blockIdx.x * 8 + wave) * 16;
    const float* arow = x + (row0 + m) * IND + 2 * kg;  // A: [16x4] frag stream
    const float* bcol = lwq + m + 2 * kg * 16;          // B: wq[k][n], n=m

    v8f c = {};
#pragma unroll 4
    for (int k0 = 0; k0 < IND; k0 += 4) {
        v2f a = *(const v2f*)(arow + k0);               // K = k0+2kg, k0+2kg+1
        v2f b;
        b.x = bcol[k0 * 16];                            // B[k0+2kg  ][n]
        b.y = bcol[k0 * 16 + 16];                       // B[k0+2kg+1][n]
        c = __builtin_amdgcn_wmma_f32_16x16x4_f32(false, a, false, b,
                                                  (short)0, c, false, false);
    }
    // C/D layout: VGPR r -> row r (lanes 0-15) or row 8+r (lanes 16-31)
    if (m < NHEADS) {
        float cqv = cq[m];
        float* orow = sw + (row0 + 8 * kg) * NHEADS + m;
#pragma unroll
        for (int r = 0; r < 8; ++r) orow[r * NHEADS] = c[r] + cqv;
    }
}

// ----------------------------- segment softmax ------------------------------
__device__ __forceinline__ int lower_bound_i32(const int* __restrict__ a,
                                               int n, int val) {
    int lo = 0, hi = n;
    while (lo < hi) {
        int mid = (lo + hi) >> 1;
        if (a[mid] < val) lo = mid + 1; else hi = mid;
    }
    return lo;
}

// One block per graph; wave h (of 8) handles head h with shuffle reductions.
__global__ void mhap_softmax(const int* __restrict__ batch,
                             float* __restrict__ sw, int n) {
    int b = blockIdx.x;
    int start = lower_bound_i32(batch, n, b);
    int end   = lower_bound_i32(batch, n, b + 1);
    if (start >= end) return;  // empty graph (probability ~0)

    int h    = threadIdx.x >> 5;   // head = wave id
    int lane = threadIdx.x & 31;

    float mx = -3.402823466e38f;
    for (int i = start + lane; i < end; i += 32)
        mx = fmaxf(mx, sw[i * NHEADS + h]);
#pragma unroll
    for (int off = 16; off > 0; off >>= 1)
        mx = fmaxf(mx, __shfl_xor(mx, off, 32));

    float dn = 0.0f;
    for (int i = start + lane; i < end; i += 32)
        dn += __expf(sw[i * NHEADS + h] - mx);
#pragma unroll
    for (int off = 16; off > 0; off >>= 1)
        dn += __shfl_xor(dn, off, 32);

    float inv = 1.0f / dn;
    for (int i = start + lane; i < end; i += 32) {
        int idx = i * NHEADS + h;
        sw[idx] = __expf(sw[idx] - mx) * inv;
    }
}

// ------------------- xw[b][h*256+i] = sum_n w[n,h] * x[n,i] -----------------
// One block per graph; thread t owns column i=t for all 8 heads (8 reg accs).
__global__ void mhap_xw(const float* __restrict__ x,
                        const float* __restrict__ w,      // [N][8] normalized
                        const int* __restrict__ batch,
                        float* __restrict__ xw, int n) {  // [B][2048]
    int b = blockIdx.x;
    int start = lower_bound_i32(batch, n, b);
    int end   = lower_bound_i32(batch, n, b + 1);
    int tid = threadIdx.x;

    __shared__ float wlds[32 * NHEADS];
    float acc[NHEADS] = {0, 0, 0, 0, 0, 0, 0, 0};

    for (int base = start; base < end; base += 32) {
        int cnt = min(32, end - base);
        __syncthreads();
        if (tid < cnt * NHEADS) wlds[tid] = w[base * NHEADS + tid];
        __syncthreads();
        for (int j = 0; j < cnt; ++j) {
            float xv = x[(long)(base + j) * IND + tid];   // coalesced 1KB/node
#pragma unroll
            for (int hh = 0; hh < NHEADS; ++hh)
                acc[hh] = fmaf(wlds[j * NHEADS + hh], xv, acc[hh]);
        }
    }
#pragma unroll
    for (int hh = 0; hh < NHEADS; ++hh)
        xw[(long)b * KCOMB + hh * IND + tid] = acc[hh];
}

// ------------- out = xw[2048x2048] @ Wcomb[2048x256] + bout (WMMA) ----------
// 2048 tiles of 16x16; 8 waves/block -> 256 blocks.
__global__ void mhap_out_gemm(const float* __restrict__ A,     // [2048][2048]
                              const float* __restrict__ Bm,    // [2048][256]
                              const float* __restrict__ bias,  // [256]
                              float* __restrict__ C) {         // [2048][256]
    const int wave = threadIdx.x >> 5;
    const int lane = threadIdx.x & 31;
    const int m    = lane & 15;
    const int kg   = lane >> 4;

    const int tile = blockIdx.x * 8 + wave;     // 0..2047
    const int tm = tile >> 4;                   // row tile 0..127
    const int tn = tile & 15;                   // col tile 0..15
    const long row0 = (long)tm * 16;
    const int col0 = tn * 16;

    const float* arow = A + (row0 + m) * KCOMB + 2 * kg;
    const float* bptr = Bm + (2 * kg) * IND + col0 + m;

    v8f c = {};
#pragma unroll 4
    for (int k0 = 0; k0 < KCOMB; k0 += 4) {
        v2f a = *(const v2f*)(arow + k0);
        v2f b;
        b.x = bptr[k0 * IND];
        b.y = bptr[k0 * IND + IND];
        c = __builtin_amdgcn_wmma_f32_16x16x4_f32(false, a, false, b,
                                                  (short)0, c, false, false);
    }
    float bvv = bias[col0 + m];
    float* crow = C + (row0 + 8 * kg) * IND + col0 + m;
#pragma unroll
    for (int r = 0; r < 8; ++r) crow[r * IND] = c[r] + bvv;
}

// ---------------------------------------------------------------------------
extern "C" void kernel_launch(void* const* d_in, const int* in_sizes, int n_in,
                              void* d_out, int out_size, void* d_ws, size_t ws_size,
                              hipStream_t stream) {
    const float* x    = (const float*)d_in[0];
    const int*   bat  = (const int*)  d_in[1];
    const float* q    = (const float*)d_in[2];
    const float* Wk   = (const float*)d_in[3];
    const float* bk   = (const float*)d_in[4];
    const float* Wv   = (const float*)d_in[5];
    const float* bv   = (const float*)d_in[6];
    const float* Wo   = (const float*)d_in[7];
    const float* bo   = (const float*)d_in[8];
    float* out = (float*)d_out;

    // workspace layout (floats)
    float* ws    = (float*)d_ws;
    float* sw    = ws;                                   // N*8      (8 MB)
    float* xw    = sw + (size_t)NATOMS * NHEADS;         // B*2048   (16 MB)
    float* wq    = xw + (size_t)NGRAPHS * KCOMB;         // 256*16
    float* cq    = wq + IND * 16;                        // 16
    float* bout  = cq + 16;                              // 256
    float* wcomb = bout + IND;                           // 2048*256 (2 MB)

    mhap_prep<<<1, 256, 0, stream>>>(q, Wk, bk, bv, Wo, bo, wq, cq, bout);
    mhap_wcomb<<<KCOMB, 256, 0, stream>>>(Wv, Wo, wcomb);
    mhap_scores<<<NATOMS / 128, 256, 0, stream>>>(x, wq, cq, sw);
    mhap_softmax<<<NGRAPHS, 256, 0, stream>>>(bat, sw, NATOMS);
    mhap_xw<<<NGRAPHS, 256, 0, stream>>>(x, sw, bat, xw, NATOMS);
    mhap_out_gemm<<<NGRAPHS * 16 / 128, 256, 0, stream>>>(xw, wcomb, bout, out);
}